// JNetwork_65137474011970
// MI455X (gfx1250) — compile-verified
//
#include <hip/hip_runtime.h>
#include <stdint.h>

// ---------------------------------------------------------------------------
// Astrochemical reaction-network RHS for MI455X (gfx1250, wave32).
//  K1: per-reaction rates, 4 reactions/thread b128 streams,
//      abundances staged to LDS via gfx1250 async-to-LDS copy
//  K2: COO scatter-add, 4 nnz/thread b128 streams, LDS-privatized ds_add_f32
//  K3: cross-workgroup reduction as ones-matmul via V_WMMA_F32_16X16X4_F32
// ---------------------------------------------------------------------------

#define NSPEC   8192
#define THREADS 256
#define NWG_SCATTER 256          // partial accumulators (8 MB in d_ws)

typedef __attribute__((ext_vector_type(2))) float v2f;
typedef __attribute__((ext_vector_type(4))) float v4f;
typedef __attribute__((ext_vector_type(8))) float v8f;
typedef __attribute__((ext_vector_type(4))) int   v4i;

#define AS1 __attribute__((address_space(1)))
#define AS3 __attribute__((address_space(3)))

// -------------------------------- K1 ---------------------------------------
__global__ __launch_bounds__(THREADS)
void rates_kernel(const float* __restrict__ ab,
                  const float* __restrict__ Tp,
                  const float* __restrict__ crp,
                  const float* __restrict__ fuvp,
                  const float* __restrict__ alpha,
                  const float* __restrict__ beta,
                  const float* __restrict__ gamma,
                  const int*   __restrict__ rtype,
                  const int*   __restrict__ rsp,     // [R,2]
                  float*       __restrict__ rates,
                  int R)
{
    __shared__ __align__(16) float s_ab[NSPEC];

    // Stage the 32KB abundance table into LDS (gfx1250 async-to-LDS path).
#if __has_builtin(__builtin_amdgcn_global_load_async_to_lds_b128)
    for (int i = threadIdx.x * 4; i < NSPEC; i += THREADS * 4) {
        __builtin_amdgcn_global_load_async_to_lds_b128(
            (AS1 v4i*)(ab + i), (AS3 v4i*)(s_ab + i), 0, 0);
    }
#if __has_builtin(__builtin_amdgcn_s_wait_asynccnt)
    __builtin_amdgcn_s_wait_asynccnt(0);
#else
    asm volatile("s_wait_asynccnt 0" ::: "memory");
#endif
#else
    for (int i = threadIdx.x; i < NSPEC; i += THREADS)
        s_ab[i] = ab[i];
#endif
    __syncthreads();

    const float T    = Tp[0];
    const float cr   = crp[0];
    const float fuv  = fuvp[0];
    const float lgT  = logf(T * (1.0f / 300.0f));
    const float invT = 1.0f / T;

    const int gid    = blockIdx.x * THREADS + threadIdx.x;
    const int stride = gridDim.x * THREADS;
    const int R4     = R >> 2;            // vec4 groups

    for (int q = gid; q < R4; q += stride) {
        const int r4 = q * 4;
        v4f a4 = __builtin_nontemporal_load((const v4f*)(alpha + r4));
        v4f b4 = __builtin_nontemporal_load((const v4f*)(beta  + r4));
        v4f g4 = __builtin_nontemporal_load((const v4f*)(gamma + r4));
        v4i t4 = __builtin_nontemporal_load((const v4i*)(rtype + r4));
        v4i sA = __builtin_nontemporal_load((const v4i*)(rsp + 2 * r4));     // r0:{s0,s1} r1:{s0,s1}
        v4i sB = __builtin_nontemporal_load((const v4i*)(rsp + 2 * r4 + 4)); // r2:{s0,s1} r3:{s0,s1}

        v4f out;
#define RATE1(j, aa, bb, gg, tt, i0, i1)                                     \
        {                                                                    \
            float k_arr = (aa) * expf((bb) * lgT - (gg) * invT);             \
            float k_cr  = (aa) * cr;                                         \
            float k_fuv = (aa) * fuv * expf(-(gg));                          \
            float k = ((tt) == 0) ? k_arr : (((tt) == 1) ? k_cr : k_fuv);    \
            out[j] = k * s_ab[i0] * s_ab[i1];                                \
        }
        RATE1(0, a4.x, b4.x, g4.x, t4.x, sA.x, sA.y)
        RATE1(1, a4.y, b4.y, g4.y, t4.y, sA.z, sA.w)
        RATE1(2, a4.z, b4.z, g4.z, t4.z, sB.x, sB.y)
        RATE1(3, a4.w, b4.w, g4.w, t4.w, sB.z, sB.w)
#undef RATE1
        __builtin_nontemporal_store(out, (v4f*)(rates + r4));
    }

    // scalar tail (dead when R % 4 == 0)
    for (int r = R4 * 4 + gid; r < R; r += stride) {
        float a = alpha[r], b = beta[r], g = gamma[r];
        int   t = rtype[r];
        float k_arr = a * expf(b * lgT - g * invT);
        float k = (t == 0) ? k_arr : ((t == 1) ? (a * cr) : (a * fuv * expf(-g)));
        rates[r] = k * s_ab[rsp[2 * r]] * s_ab[rsp[2 * r + 1]];
    }
}

// -------------------------------- K2 ---------------------------------------
__global__ __launch_bounds__(THREADS)
void scatter_kernel(const int*   __restrict__ rows,
                    const int*   __restrict__ cols,
                    const float* __restrict__ vals,
                    const float* __restrict__ rates,
                    float*       __restrict__ partials,   // [NWG][NSPEC]
                    int nnz)
{
    __shared__ float acc[NSPEC];
    for (int i = threadIdx.x; i < NSPEC; i += THREADS) acc[i] = 0.0f;
    __syncthreads();

    const int nv    = nnz >> 2;                           // vec4 groups
    const int per   = (nv + gridDim.x - 1) / gridDim.x;
    const int begin = blockIdx.x * per;
    const int end   = (begin + per < nv) ? (begin + per) : nv;

    const v4i* rows4 = (const v4i*)rows;
    const v4i* cols4 = (const v4i*)cols;
    const v4f* vals4 = (const v4f*)vals;

    for (int i = begin + threadIdx.x; i < end; i += THREADS) {
        v4i r4 = __builtin_nontemporal_load(rows4 + i);
        v4i c4 = __builtin_nontemporal_load(cols4 + i);
        v4f v4 = __builtin_nontemporal_load(vals4 + i);
        atomicAdd(&acc[r4.x], v4.x * rates[c4.x]);        // ds_add_f32
        atomicAdd(&acc[r4.y], v4.y * rates[c4.y]);
        atomicAdd(&acc[r4.z], v4.z * rates[c4.z]);
        atomicAdd(&acc[r4.w], v4.w * rates[c4.w]);
    }

    // scalar tail (dead when nnz % 4 == 0), done by last block
    if (blockIdx.x == gridDim.x - 1) {
        for (int i = nv * 4 + threadIdx.x; i < nnz; i += THREADS)
            atomicAdd(&acc[rows[i]], vals[i] * rates[cols[i]]);
    }
    __syncthreads();

    float* dst = partials + (size_t)blockIdx.x * NSPEC;
    for (int i = threadIdx.x; i < NSPEC; i += THREADS)
        dst[i] = acc[i];
}

// -------------------------------- K3 ---------------------------------------
// d_out[s] = sum_w partials[w][s] as ones[16x4] x partials[4x16] WMMA chain.
// A = all-ones, so the column sum is invariant to B's K-striping; all D rows
// are equal, so reading D row 0 (c[0], lanes 0..15) is safe.
__global__ __launch_bounds__(128)
void reduce_wmma_kernel(const float* __restrict__ partials,
                        float*       __restrict__ out,
                        int nparts)
{
    const int wave  = (blockIdx.x * blockDim.x + threadIdx.x) >> 5;
    const int lane  = threadIdx.x & 31;
    const int col   = wave * 16 + (lane & 15);       // species column
    const int khalf = (lane >> 4) & 1;               // 0: lanes 0-15, 1: 16-31

    v2f a; a.x = 1.0f; a.y = 1.0f;                   // ones A (layout-invariant)
    v8f c = {};

#if __has_builtin(__builtin_amdgcn_wmma_f32_16x16x4_f32)
    for (int w = 0; w < nparts; w += 4) {
        v2f b;
        b.x = partials[(size_t)(w + 2 * khalf    ) * NSPEC + col];
        b.y = partials[(size_t)(w + 2 * khalf + 1) * NSPEC + col];
        c = __builtin_amdgcn_wmma_f32_16x16x4_f32(
                false, a, false, b, (short)0, c, false, false);
    }
    if (lane < 16) out[col] = c[0];
#else
    float s = 0.0f;
    for (int w = khalf; w < nparts; w += 2)
        s += partials[(size_t)w * NSPEC + col];
    __shared__ float red[128];
    red[threadIdx.x] = s;
    __syncthreads();
    if (lane < 16) out[col] = red[threadIdx.x] + red[threadIdx.x + 16];
#endif
}

// ------------------------------ launcher -----------------------------------
extern "C" void kernel_launch(void* const* d_in, const int* in_sizes, int n_in,
                              void* d_out, int out_size, void* d_ws, size_t ws_size,
                              hipStream_t stream)
{
    const float* ab    = (const float*)d_in[0];
    const float* T     = (const float*)d_in[1];
    const float* cr    = (const float*)d_in[2];
    const float* fuv   = (const float*)d_in[3];
    const float* alpha = (const float*)d_in[4];
    const float* beta  = (const float*)d_in[5];
    const float* gamma = (const float*)d_in[6];
    const int*   rtype = (const int*)d_in[7];
    const int*   rsp   = (const int*)d_in[8];
    const int*   rows  = (const int*)d_in[9];
    const int*   cols  = (const int*)d_in[10];
    const float* vals  = (const float*)d_in[11];

    const int R   = in_sizes[4];   // 2,097,152
    const int nnz = in_sizes[9];   // 8,388,608

    // d_ws layout: [0, R) rates | [R, R + NWG*NSPEC) partials  (~16 MB)
    float* rates    = (float*)d_ws;
    float* partials = rates + R;

    rates_kernel<<<512, THREADS, 0, stream>>>(
        ab, T, cr, fuv, alpha, beta, gamma, rtype, rsp, rates, R);

    scatter_kernel<<<NWG_SCATTER, THREADS, 0, stream>>>(
        rows, cols, vals, rates, partials, nnz);

    const int tiles = NSPEC / 16;                 // 512 waves, 32 lanes each
    reduce_wmma_kernel<<<tiles * 32 / 128, 128, 0, stream>>>(
        partials, (float*)d_out, NWG_SCATTER);
}